// GCN_50139448213989
// MI455X (gfx1250) — compile-verified
//
#include <hip/hip_runtime.h>
#include <hip/hip_bf16.h>

typedef float v2f __attribute__((ext_vector_type(2)));
typedef float v8f __attribute__((ext_vector_type(8)));

#define HFEAT 96

// ---------------------------------------------------------------------------
// Dense GEMM: Y[n x 96] = X[n x 96] @ W^T (W is [96 x 96] row-major)
// One wave computes one 16x16 tile with fp32 WMMA (16x16x4), K-loop = 24 steps.
// A layout (32-bit 16x4): v0 = K=k0 (lanes 0-15) / K=k0+2 (lanes 16-31),
//                         v1 = K=k0+1 / k0+3  -> a single float2 load per lane.
// B mirrors A with N striped across lanes: B[k][n] = W[n][k].
// C/D: 8 VGPRs, VGPR j holds row m0+j (lanes 0-15) / m0+j+8 (lanes 16-31).
// ---------------------------------------------------------------------------
__global__ __launch_bounds__(128)
void gcn_gemm_wmma_f32(const float* __restrict__ X, const float* __restrict__ Wm,
                       float* __restrict__ Y, int n) {
  const int wave = threadIdx.x >> 5;
  const int lane = threadIdx.x & 31;
  const int half = lane >> 4;     // 0: lanes 0-15, 1: lanes 16-31
  const int l    = lane & 15;

  const int mtiles = (n + 15) >> 4;
  const int mtile  = blockIdx.y * 4 + wave;   // 4 waves per block
  if (mtile >= mtiles) return;                // wave-uniform: EXEC stays all-1s
  const int m0 = mtile << 4;
  const int n0 = blockIdx.x << 4;             // 6 N-tiles for H=96

  int arow = m0 + l; if (arow >= n) arow = n - 1;       // clamp tail rows
  const float* aptr = X  + (size_t)arow * HFEAT + 2 * half;
  const float* bptr = Wm + (size_t)(n0 + l) * HFEAT + 2 * half;

  v8f acc = {};
#pragma unroll
  for (int k0 = 0; k0 < HFEAT; k0 += 4) {
    v2f a = *(const v2f*)(aptr + k0);
    v2f b = *(const v2f*)(bptr + k0);
    // 8 args: (neg_a, A, neg_b, B, c_mod, C, reuse_a, reuse_b)
    acc = __builtin_amdgcn_wmma_f32_16x16x4_f32(false, a, false, b,
                                                (short)0, acc, false, false);
  }

#pragma unroll
  for (int j = 0; j < 8; ++j) {
    int r = m0 + j + 8 * half;
    if (r < n) Y[(size_t)r * HFEAT + n0 + l] = acc[j];
  }
}

// ---------------------------------------------------------------------------
// Degree / normalization precompute
// ---------------------------------------------------------------------------
__global__ void gcn_deg_init(float* __restrict__ deg, int n) {
  int i = blockIdx.x * blockDim.x + threadIdx.x;
  if (i < n) deg[i] = 1.0f;                      // self-loop weight
}

__global__ void gcn_deg_acc(const int* __restrict__ col, const float* __restrict__ ew,
                            float* __restrict__ deg, int E) {
  int e = blockIdx.x * blockDim.x + threadIdx.x;
  if (e < E) atomicAdd(&deg[col[e]], ew[e]);
}

__global__ void gcn_dinv(float* __restrict__ deg, int n) {
  int i = blockIdx.x * blockDim.x + threadIdx.x;
  if (i < n) {
    float d = deg[i];
    deg[i] = (d > 0.0f) ? rsqrtf(d) : 0.0f;      // in-place: deg becomes dinv
  }
}

// ---------------------------------------------------------------------------
// agg[i][f] = dinv[i]^2 * xt[i][f] + b[f]   (self-loop + bias)
// ---------------------------------------------------------------------------
__global__ void gcn_agg_init(const float* __restrict__ xt, const float* __restrict__ dinv,
                             const float* __restrict__ b, float* __restrict__ agg, int n) {
  int idx = blockIdx.x * blockDim.x + threadIdx.x;
  int total = n * HFEAT;
  if (idx < total) {
    int i = idx / HFEAT;
    int f = idx - i * HFEAT;
    float di = dinv[i];
    agg[idx] = di * di * xt[idx] + b[f];
  }
}

// ---------------------------------------------------------------------------
// Edge scatter: one wave per edge; each lane handles features f, f+32, f+64.
// agg[col] += dinv[row]*ew*dinv[col] * xt[row]   (L2-resident atomics)
// ---------------------------------------------------------------------------
__global__ __launch_bounds__(256)
void gcn_scatter(const int* __restrict__ row, const int* __restrict__ col,
                 const float* __restrict__ ew, const float* __restrict__ dinv,
                 const float* __restrict__ xt, float* __restrict__ agg, int E) {
  const int wpb   = blockDim.x >> 5;
  const int wid   = blockIdx.x * wpb + (threadIdx.x >> 5);
  const int lane  = threadIdx.x & 31;
  const int nwave = gridDim.x * wpb;
  for (int e = wid; e < E; e += nwave) {
    int r = row[e], c = col[e];
    float nrm = dinv[r] * ew[e] * dinv[c];
    const float* src = xt  + (size_t)r * HFEAT;
    float*       dst = agg + (size_t)c * HFEAT;
#pragma unroll
    for (int f = lane; f < HFEAT; f += 32)
      atomicAdd(dst + f, nrm * src[f]);
  }
}

__global__ void gcn_relu(float* __restrict__ a, int total) {
  int i = blockIdx.x * blockDim.x + threadIdx.x;
  if (i < total) a[i] = fmaxf(a[i], 0.0f);
}

// ---------------------------------------------------------------------------
// Final FC fused with ReLU: out[i] = bfc + sum_f max(h[i][f],0) * Wfc[f]
// One wave32 per node; shuffle reduction.
// ---------------------------------------------------------------------------
__global__ __launch_bounds__(256)
void gcn_fc(const float* __restrict__ H2, const float* __restrict__ Wfc,
            const float* __restrict__ bfc, float* __restrict__ out, int n) {
  const int wid  = blockIdx.x * (blockDim.x >> 5) + (threadIdx.x >> 5);
  const int lane = threadIdx.x & 31;
  if (wid >= n) return;
  const float* h = H2 + (size_t)wid * HFEAT;
  float s = 0.0f;
#pragma unroll
  for (int f = lane; f < HFEAT; f += 32) {
    float v = fmaxf(h[f], 0.0f);
    s += v * Wfc[f];
  }
#pragma unroll
  for (int off = 16; off > 0; off >>= 1) s += __shfl_xor(s, off, 32);
  if (lane == 0) out[wid] = s + bfc[0];
}

// ---------------------------------------------------------------------------
extern "C" void kernel_launch(void* const* d_in, const int* in_sizes, int n_in,
                              void* d_out, int out_size, void* d_ws, size_t ws_size,
                              hipStream_t stream) {
  const float* x   = (const float*)d_in[0];
  const int*   ei  = (const int*)  d_in[1];
  const float* ew  = (const float*)d_in[2];
  const float* W1  = (const float*)d_in[3];
  const float* b1  = (const float*)d_in[4];
  const float* W2  = (const float*)d_in[5];
  const float* b2  = (const float*)d_in[6];
  const float* Wfc = (const float*)d_in[7];
  const float* bfc = (const float*)d_in[8];

  const int n = in_sizes[0] / HFEAT;
  const int E = in_sizes[2];
  const int* row = ei;
  const int* col = ei + E;

  // Workspace partition: deg/dinv [n] | xt [n*96] | agg [n*96]
  float* deg = (float*)d_ws;
  float* xt  = deg + n;
  float* agg = xt + (size_t)n * HFEAT;
  float* out = (float*)d_out;

  const int total = n * HFEAT;
  const int mtiles = (n + 15) >> 4;
  dim3 gemmGrid(HFEAT / 16, (mtiles + 3) / 4);
  const int scatterBlocks = (E + 7) / 8;         // 8 waves (edges) per 256-thr block
  const int fcBlocks = (n + 7) / 8;

  // --- normalization precompute (shared by both layers) ---
  gcn_deg_init<<<(n + 255) / 256, 256, 0, stream>>>(deg, n);
  gcn_deg_acc <<<(E + 255) / 256, 256, 0, stream>>>(col, ew, deg, E);
  gcn_dinv    <<<(n + 255) / 256, 256, 0, stream>>>(deg, n);   // deg -> dinv

  // --- layer 1 ---
  gcn_gemm_wmma_f32<<<gemmGrid, 128, 0, stream>>>(x, W1, xt, n);
  gcn_agg_init<<<(total + 255) / 256, 256, 0, stream>>>(xt, deg, b1, agg, n);
  gcn_scatter <<<scatterBlocks, 256, 0, stream>>>(row, col, ew, deg, xt, agg, E);
  gcn_relu    <<<(total + 255) / 256, 256, 0, stream>>>(agg, total);

  // --- layer 2 (xt buffer reused for xt2, agg reused for agg2) ---
  gcn_gemm_wmma_f32<<<gemmGrid, 128, 0, stream>>>(agg, W2, xt, n);
  gcn_agg_init<<<(total + 255) / 256, 256, 0, stream>>>(xt, deg, b2, agg, n);
  gcn_scatter <<<scatterBlocks, 256, 0, stream>>>(row, col, ew, deg, xt, agg, E);

  // --- fused ReLU + FC ---
  gcn_fc<<<fcBlocks, 256, 0, stream>>>(agg, Wfc, bfc, out, n);
}